// ImprovedVectorQuantizer_7507602833767
// MI455X (gfx1250) — compile-verified
//
#include <hip/hip_runtime.h>
#include <math.h>

typedef float v2f __attribute__((ext_vector_type(2)));
typedef float v8f __attribute__((ext_vector_type(8)));

#define N_TOT   65536      // B*H*W vectors
#define E_DIM   64
#define N_E     1024
#define HW      4096       // H*W
#define Q_ELEMS 4194304    // 16*64*64*64
#define OFF_LOSS 0
#define OFF_Q    1
#define OFF_P    (1 + Q_ELEMS)          // 4194305
#define OFF_ENC  (2 + Q_ELEMS)          // 4194306 (8B-aligned in floats, not 16B)

// ---------------------------------------------------------------- prep ------
__global__ void vq_prep(const float* __restrict__ embed, float* __restrict__ e2,
                        int* __restrict__ hist, float* __restrict__ partial) {
    int j = blockIdx.x * 256 + threadIdx.x;
    if (j < N_E) {
        const float* row = embed + (size_t)j * E_DIM;
        float s = 0.f;
        #pragma unroll
        for (int c = 0; c < E_DIM; ++c) { float v = row[c]; s += v * v; }
        e2[j]   = s;
        hist[j] = 0;
    }
    if (blockIdx.x == 0 && threadIdx.x < 256) partial[threadIdx.x] = 0.f;
}

// ------------------------------------------------------------- argmin ------
// Each wave: 16 rows x all 1024 codes via v_wmma_f32_16x16x4_f32.
// dist surrogate = ||e||^2 - 2 x.e  (||x||^2 constant per row, dropped).
__launch_bounds__(256)
__global__ void vq_argmin(const float* __restrict__ inp,
                          const float* __restrict__ embed,
                          const float* __restrict__ e2g,
                          int* __restrict__ idxg,
                          int* __restrict__ histg) {
    __shared__ float s_emb[128 * 68];   // 128 codes x 64 dims, stride 68 (bank-safe, 16B-aligned rows)
    __shared__ float s_e2[N_E];
    __shared__ int   s_hist[N_E];

    const int tid   = threadIdx.x;
    const int lane  = tid & 31;
    const int wave  = tid >> 5;
    const int half  = lane >> 4;        // 0: lanes 0-15, 1: lanes 16-31
    const int mlo   = lane & 15;
    const int khalf = half * 2;         // A/B fragment K-offset per ISA 16x4 layout

    #pragma unroll
    for (int i = 0; i < 4; ++i) {
        s_e2[tid + i * 256]   = e2g[tid + i * 256];
        s_hist[tid + i * 256] = 0;
    }

    // A fragments: this wave's 16 rows, all K=64 (strided-channel reads, lane-coalesced over hw)
    const int r0 = (blockIdx.x * 8 + wave) * 16;
    const int n  = r0 + mlo;
    const int bb = n >> 12;
    const int hw = n & (HW - 1);
    v2f afrag[16];
    #pragma unroll
    for (int kk = 0; kk < 16; ++kk) {
        int k0 = kk * 4 + khalf;
        const float* p = inp + (size_t)(bb * E_DIM + k0) * HW + hw;
        afrag[kk].x = p[0];
        afrag[kk].y = p[HW];
    }

    float best_d[8];
    int   best_i[8];
    #pragma unroll
    for (int v = 0; v < 8; ++v) { best_d[v] = 3.4e38f; best_i[v] = 0; }

    for (int chunk = 0; chunk < 8; ++chunk) {
        __syncthreads();
        // cooperative stage of 128 codes (float4 global + float4 LDS, both 16B aligned)
        #pragma unroll
        for (int it = 0; it < 8; ++it) {
            int e    = it * 256 + tid;      // 0..2047
            int code = e >> 4;              // 0..127
            int kg   = (e & 15) * 4;        // 0,4,...,60
            float4 v4 = *(const float4*)(embed + (size_t)(chunk * 128 + code) * E_DIM + kg);
            *(float4*)(&s_emb[code * 68 + kg]) = v4;
        }
        __syncthreads();

        #pragma unroll 1
        for (int t = 0; t < 8; ++t) {
            const int codeBase = chunk * 128 + t * 16;
            const float* brow  = &s_emb[(t * 16 + mlo) * 68 + khalf];
            v8f acc = {};
            #pragma unroll
            for (int kk = 0; kk < 16; ++kk) {
                v2f bfrag = *(const v2f*)(brow + kk * 4);
                acc = __builtin_amdgcn_wmma_f32_16x16x4_f32(
                          false, afrag[kk], false, bfrag, (short)0, acc, false, false);
            }
            const float e2v = s_e2[codeBase + mlo];
            const int   myi = codeBase + mlo;
            #pragma unroll
            for (int v = 0; v < 8; ++v) {
                // D layout: lane holds (M = v + 8*half, N = lane&15)
                float d = e2v - 2.0f * acc[v];
                int   i = myi;
                #pragma unroll
                for (int s = 1; s < 16; s <<= 1) {   // min+argmin within each 16-lane half
                    float od = __shfl_xor(d, s, 32);
                    int   oi = __shfl_xor(i, s, 32);
                    if (od < d || (od == d && oi < i)) { d = od; i = oi; }
                }
                if (d < best_d[v] || (d == best_d[v] && i < best_i[v])) {
                    best_d[v] = d; best_i[v] = i;
                }
            }
        }
    }

    // lane 0 owns rows r0..r0+7, lane 16 owns rows r0+8..r0+15
    if (mlo == 0) {
        #pragma unroll
        for (int v = 0; v < 8; ++v) {
            int row = r0 + v + 8 * half;
            idxg[row] = best_i[v];
            atomicAdd(&s_hist[best_i[v]], 1);
        }
    }
    __syncthreads();
    #pragma unroll
    for (int i = 0; i < 4; ++i) {
        int c = s_hist[tid + i * 256];
        if (c) atomicAdd(&histg[tid + i * 256], c);
    }
}

// ------------------------------------------------- zero-fill one-hot -------
__global__ void vq_fill_enc(float* __restrict__ out) {
    size_t g = (size_t)blockIdx.x * 256 + threadIdx.x;  // 0 .. 8388607, 8 floats each
    float2* p = (float2*)(out + OFF_ENC) + g * 4;
    float2 z; z.x = 0.f; z.y = 0.f;
    p[0] = z; p[1] = z; p[2] = z; p[3] = z;
}

// ------------------------------- gather / transpose / scatter / loss -------
__global__ void vq_out(const float* __restrict__ inp, const float* __restrict__ embed,
                       const int* __restrict__ idxg, float* __restrict__ out,
                       float* __restrict__ partial) {
    __shared__ float red[256];
    int n  = blockIdx.x * 256 + threadIdx.x;   // 0..65535
    int i  = idxg[n];
    int bb = n >> 12;
    int hw = n & (HW - 1);
    const float* e = embed + (size_t)i * E_DIM;
    float s = 0.f;
    #pragma unroll
    for (int c = 0; c < E_DIM; ++c) {
        float  q   = e[c];
        size_t off = (size_t)(bb * E_DIM + c) * HW + hw;   // (B,C,H,W) layout
        float  x   = inp[off];
        float  dd  = q - x;
        s += dd * dd;
        out[OFF_Q + off] = q;                  // straight-through value == q
    }
    out[(size_t)OFF_ENC + (size_t)n * N_E + i] = 1.0f;     // one-hot

    red[threadIdx.x] = s;
    __syncthreads();
    for (int st = 128; st > 0; st >>= 1) {
        if (threadIdx.x < st) red[threadIdx.x] += red[threadIdx.x + st];
        __syncthreads();
    }
    if (threadIdx.x == 0) partial[blockIdx.x] = red[0];
}

// ----------------------------------------------------------- finalize ------
__global__ void vq_final(const float* __restrict__ partial, const int* __restrict__ hist,
                         float* __restrict__ out) {
    __shared__ float red[256];
    int tid = threadIdx.x;
    float s = partial[tid];
    float h = 0.f;
    #pragma unroll
    for (int i = 0; i < 4; ++i) {
        float p = (float)hist[tid + i * 256] * (1.0f / 65536.0f);
        h += p * logf(p + 1e-10f);
    }
    red[tid] = s;
    __syncthreads();
    for (int st = 128; st > 0; st >>= 1) { if (tid < st) red[tid] += red[tid + st]; __syncthreads(); }
    float S = red[0];
    __syncthreads();
    red[tid] = h;
    __syncthreads();
    for (int st = 128; st > 0; st >>= 1) { if (tid < st) red[tid] += red[tid + st]; __syncthreads(); }
    if (tid == 0) {
        out[OFF_LOSS] = 1.25f * S * (1.0f / (float)Q_ELEMS);  // q_loss + beta*e_loss, values equal
        out[OFF_P]    = expf(-red[0]);                        // perplexity
    }
}

// -------------------------------------------------------------- launch -----
extern "C" void kernel_launch(void* const* d_in, const int* in_sizes, int n_in,
                              void* d_out, int out_size, void* d_ws, size_t ws_size,
                              hipStream_t stream) {
    const float* inp   = (const float*)d_in[0];   // (16,64,64,64) f32
    const float* embed = (const float*)d_in[1];   // (1024,64) f32
    float* out = (float*)d_out;

    float* e2      = (float*)d_ws;                // 1024 f32
    float* partial = e2 + N_E;                    // 256 f32
    int*   idx     = (int*)(partial + 256);       // 65536 i32
    int*   hist    = idx + N_TOT;                 // 1024 i32

    vq_prep   <<<4,     256, 0, stream>>>(embed, e2, hist, partial);
    vq_argmin <<<512,   256, 0, stream>>>(inp, embed, e2, idx, hist);
    vq_fill_enc<<<32768,256, 0, stream>>>(out);
    vq_out    <<<256,   256, 0, stream>>>(inp, embed, idx, out, partial);
    vq_final  <<<1,     256, 0, stream>>>(partial, hist, out);
}